// NormalizedHistogram_23888608100752
// MI455X (gfx1250) — compile-verified
//
#include <hip/hip_runtime.h>
#include <hip/hip_bf16.h>

typedef float v4f __attribute__((ext_vector_type(4)));
typedef int   v4i __attribute__((ext_vector_type(4)));

#define NBINS      256
#define TPB        256
#define NWAVES     8            // TPB / wave32
#define K_UNROLL   4            // float4 loads per thread per tile
#define TILE4      (TPB * K_UNROLL)   // 1024 float4 = 4096 floats per tile
#define MAX_BLOCKS 2048

#if defined(__has_builtin)
#  if __has_builtin(__builtin_amdgcn_global_load_async_to_lds_b128) && \
      __has_builtin(__builtin_amdgcn_s_wait_asynccnt)
#    define USE_ASYNC_LDS 1
#  endif
#endif

// TF histogram_fixed_width binning over [0,1]: idx = clip((int)(x*256), 0, 255)
__device__ __forceinline__ void accum_elem(unsigned e, float x, unsigned* h) {
  int b = (int)(x * 256.0f);          // C truncation == astype(int32) toward zero
  b = b < 0 ? 0 : b;
  b = b > (NBINS - 1) ? (NBINS - 1) : b;
  unsigned c = e % 3u;                // channel is fastest-varying dim
  atomicAdd(&h[c * NBINS + (unsigned)b], 1u);   // -> ds_add_u32 (no return)
}

__global__ __launch_bounds__(TPB) void hist_main(const float* __restrict__ in,
                                                 int n,
                                                 unsigned* __restrict__ gbins) {
  __shared__ unsigned hist[NWAVES][3 * NBINS];        // 24 KB, per-wave privatized
#if USE_ASYNC_LDS
  __shared__ v4f stage[2][K_UNROLL][TPB];             // 32 KB double-buffered staging
#endif
  const unsigned tid = threadIdx.x;
  const unsigned wid = tid >> 5;                      // wave32
  unsigned* myh = hist[wid];

  for (unsigned i = tid; i < NWAVES * 3 * NBINS; i += TPB) (&hist[0][0])[i] = 0u;
  __syncthreads();

  const unsigned n4       = (unsigned)n >> 2;
  const unsigned numTiles = (n4 + TILE4 - 1) / TILE4;
  const v4f* in4 = (const v4f*)in;

#if USE_ASYNC_LDS
  // --- Async global->LDS double-buffered streaming (gfx1250 ASYNCcnt path) ---
  unsigned t   = blockIdx.x;
  unsigned buf = 0;
  if (t < numTiles) {
#pragma unroll
    for (int k = 0; k < K_UNROLL; ++k) {
      unsigned g = t * TILE4 + (unsigned)k * TPB + tid;
      if (g >= n4) g = n4 - 1;                        // clamp: keep EXEC uniform
      __builtin_amdgcn_global_load_async_to_lds_b128(
          (__attribute__((address_space(1))) v4i*)(in4 + g),
          (__attribute__((address_space(3))) v4i*)(&stage[buf][k][tid]),
          0, 0);
    }
  }
  while (t < numTiles) {
    unsigned tn = t + gridDim.x;
    if (tn < numTiles) {
#pragma unroll
      for (int k = 0; k < K_UNROLL; ++k) {
        unsigned g = tn * TILE4 + (unsigned)k * TPB + tid;
        if (g >= n4) g = n4 - 1;
        __builtin_amdgcn_global_load_async_to_lds_b128(
            (__attribute__((address_space(1))) v4i*)(in4 + g),
            (__attribute__((address_space(3))) v4i*)(&stage[buf ^ 1][k][tid]),
            0, 0);
      }
      __builtin_amdgcn_s_wait_asynccnt(K_UNROLL);     // current tile complete
    } else {
      __builtin_amdgcn_s_wait_asynccnt(0);
    }
#pragma unroll
    for (int k = 0; k < K_UNROLL; ++k) {
      unsigned g = t * TILE4 + (unsigned)k * TPB + tid;
      if (g < n4) {
        v4f v = stage[buf][k][tid];                   // lane reads only its own slot
        unsigned e = g * 4u;
        accum_elem(e + 0u, v.x, myh);
        accum_elem(e + 1u, v.y, myh);
        accum_elem(e + 2u, v.z, myh);
        accum_elem(e + 3u, v.w, myh);
      }
    }
    buf ^= 1;
    t = tn;
  }
#else
  // --- Fallback: nontemporal 128-bit vector loads straight to VGPRs ---
  for (unsigned t = blockIdx.x; t < numTiles; t += gridDim.x) {
#pragma unroll
    for (int k = 0; k < K_UNROLL; ++k) {
      unsigned g = t * TILE4 + (unsigned)k * TPB + tid;
      if (g < n4) {
        v4f v = __builtin_nontemporal_load(in4 + g);
        unsigned e = g * 4u;
        accum_elem(e + 0u, v.x, myh);
        accum_elem(e + 1u, v.y, myh);
        accum_elem(e + 2u, v.z, myh);
        accum_elem(e + 3u, v.w, myh);
      }
    }
  }
#endif

  // tail (n % 4) handled by block 0
  unsigned rem = (unsigned)n & 3u;
  if (blockIdx.x == 0 && tid < rem) {
    unsigned e = n4 * 4u + tid;
    accum_elem(e, in[e], myh);
  }

  __syncthreads();
  // merge the 8 per-wave copies, push to global counters
  for (unsigned b = tid; b < 3 * NBINS; b += TPB) {
    unsigned s = 0;
#pragma unroll
    for (int w = 0; w < NWAVES; ++w) s += hist[w][b];
    if (s) atomicAdd(&gbins[b], s);
  }
}

__global__ void zero_bins(unsigned* __restrict__ b) {
  unsigned i = blockIdx.x * blockDim.x + threadIdx.x;
  if (i < 3 * NBINS) b[i] = 0u;
}

__global__ __launch_bounds__(NBINS) void finalize(const unsigned* __restrict__ bins,
                                                  float* __restrict__ out) {
  __shared__ unsigned s[3][NBINS];
  const unsigned b = threadIdx.x;
  unsigned c0 = bins[0 * NBINS + b];
  unsigned c1 = bins[1 * NBINS + b];
  unsigned c2 = bins[2 * NBINS + b];
  s[0][b] = c0; s[1][b] = c1; s[2][b] = c2;
  __syncthreads();
  for (unsigned st = NBINS / 2; st > 0; st >>= 1) {
    if (b < st) {
      s[0][b] += s[0][b + st];
      s[1][b] += s[1][b + st];
      s[2][b] += s[2][b + st];
    }
    __syncthreads();
  }
  float sum0 = (float)s[0][0], sum1 = (float)s[1][0], sum2 = (float)s[2][0];
  // output layout = hist.T : [nbins, 3] row-major -> out[bin*3 + c]
  out[b * 3u + 0u] = (float)c0 / sum0;
  out[b * 3u + 1u] = (float)c1 / sum1;
  out[b * 3u + 2u] = (float)c2 / sum2;
}

extern "C" void kernel_launch(void* const* d_in, const int* in_sizes, int n_in,
                              void* d_out, int out_size, void* d_ws, size_t ws_size,
                              hipStream_t stream) {
  (void)n_in; (void)out_size; (void)ws_size;
  const float* in = (const float*)d_in[0];
  int n = in_sizes[0];
  unsigned* bins = (unsigned*)d_ws;       // 768 * 4 B of scratch
  float* out = (float*)d_out;

  zero_bins<<<3, 256, 0, stream>>>(bins);

  unsigned n4       = (unsigned)n >> 2;
  unsigned numTiles = (n4 + TILE4 - 1) / TILE4;
  unsigned blocks   = numTiles < MAX_BLOCKS ? (numTiles ? numTiles : 1u) : (unsigned)MAX_BLOCKS;

  hist_main<<<dim3(blocks), dim3(TPB), 0, stream>>>(in, n, bins);
  finalize<<<1, NBINS, 0, stream>>>(bins, out);
}